// neigh_Conv_30434138259616
// MI455X (gfx1250) — compile-verified
//
#include <hip/hip_runtime.h>
#include <stdint.h>

#define CHANNEL  200
#define NEIGH    8
#define OUT_C    193               // CHANNEL - NEIGH + 1
#define HW       16384             // 128*128
#define NBATCH   16
#define NCHUNK   2
#define TPB      256
#define W_ELEMS  (OUT_C * NEIGH)   // 1544
#define SM_ELEMS (W_ELEMS + OUT_C) // 1737 floats = 6948 B LDS

#if defined(__AMDGCN__) && __has_builtin(__builtin_amdgcn_global_load_async_to_lds_b32)
#define HAVE_ASYNC_LDS 1
#else
#define HAVE_ASYNC_LDS 0
#endif

__global__ __launch_bounds__(TPB) void neigh_conv_kernel(
    const float* __restrict__ x, const float* __restrict__ W,
    const float* __restrict__ b, float* __restrict__ out)
{
    __shared__ float sm[SM_ELEMS];

    // ---- Stage W (193x8) and b (193) into LDS via CDNA5 async-to-LDS path ----
#if HAVE_ASYNC_LDS
    for (int i = threadIdx.x; i < W_ELEMS; i += TPB)
        __builtin_amdgcn_global_load_async_to_lds_b32(
            (int*)(W + i), (int*)(sm + i), 0, 0);
    for (int i = threadIdx.x; i < OUT_C; i += TPB)
        __builtin_amdgcn_global_load_async_to_lds_b32(
            (int*)(b + i), (int*)(sm + W_ELEMS + i), 0, 0);
#if __has_builtin(__builtin_amdgcn_s_wait_asynccnt)
    __builtin_amdgcn_s_wait_asynccnt(0);
#else
    asm volatile("s_wait_asynccnt 0" ::: "memory");
#endif
#else
    for (int i = threadIdx.x; i < W_ELEMS; i += TPB) sm[i] = W[i];
    for (int i = threadIdx.x; i < OUT_C;   i += TPB) sm[W_ELEMS + i] = b[i];
#endif
    __syncthreads();

    // ---- Decode block: bid = ((pb*16 + n)*2 + chunk) ----
    const int bid   = blockIdx.x;
    const int chunk = bid & (NCHUNK - 1);
    const int rest  = bid >> 1;
    const int n     = rest & (NBATCH - 1);
    const int pb    = rest >> 4;                       // 0..15
    const int p     = (pb * TPB + threadIdx.x) * 4;    // float4 pixel offset

    const int c0 = chunk ? 96 : 0;
    const int c1 = chunk ? OUT_C : 96;                 // 96 or 193

    const float* xcol = x   + ((size_t)n * CHANNEL) * HW + p;
    float*       ocol = out + ((size_t)n * OUT_C)   * HW + p;

    // ---- Sliding channel window: win[j] = x[c + j] (8 x float4 in VGPRs) ----
    float4 win[NEIGH];
#pragma unroll
    for (int j = 0; j < NEIGH; ++j)
        win[j] = *(const float4*)(xcol + (size_t)(c0 + j) * HW);

    int c = c0;
#pragma unroll 1
    for (int g = 0; g < 12; ++g) {                     // 12 groups of 8 channels
        // Prefetch next group's planes (64 KB stride defeats HW prefetch).
        int pc = c + 16;
        if (pc > CHANNEL - NEIGH) pc = CHANNEL - NEIGH;
#pragma unroll
        for (int j = 0; j < NEIGH; ++j)
            __builtin_prefetch(xcol + (size_t)(pc + j) * HW, 0, 3);

#pragma unroll
        for (int j = 0; j < NEIGH; ++j) {
            const int cc = c + j;
            const float* wr = &sm[cc * NEIGH];
            const float bias = sm[W_ELEMS + cc];
            float4 acc = make_float4(bias, bias, bias, bias);
#pragma unroll
            for (int k = 0; k < NEIGH; ++k) {
                const float  wv = wr[k];
                const float4 xv = win[(j + k) & 7];    // static index after unroll
                acc.x = fmaf(wv, xv.x, acc.x);
                acc.y = fmaf(wv, xv.y, acc.y);
                acc.z = fmaf(wv, xv.z, acc.z);
                acc.w = fmaf(wv, xv.w, acc.w);
            }
            *(float4*)(ocol + (size_t)cc * HW) = acc;
            // Slide: slot j now holds x[cc + 8]. Max index 199 (in bounds).
            win[j] = *(const float4*)(xcol + (size_t)(cc + NEIGH) * HW);
        }
        c += NEIGH;
    }

    // ---- Remainder channel (only chunk 1: c == 192). win[j] == x[192+j]. ----
    if (c < c1) {
        const float* wr = &sm[c * NEIGH];
        const float bias = sm[W_ELEMS + c];
        float4 acc = make_float4(bias, bias, bias, bias);
#pragma unroll
        for (int k = 0; k < NEIGH; ++k) {
            const float  wv = wr[k];
            const float4 xv = win[k];
            acc.x = fmaf(wv, xv.x, acc.x);
            acc.y = fmaf(wv, xv.y, acc.y);
            acc.z = fmaf(wv, xv.z, acc.z);
            acc.w = fmaf(wv, xv.w, acc.w);
        }
        *(float4*)(ocol + (size_t)c * HW) = acc;
    }
}

extern "C" void kernel_launch(void* const* d_in, const int* in_sizes, int n_in,
                              void* d_out, int out_size, void* d_ws, size_t ws_size,
                              hipStream_t stream) {
    const float* x = (const float*)d_in[0];   // (16, 200, 128, 128)
    const float* W = (const float*)d_in[1];   // (193, 8)
    const float* b = (const float*)d_in[2];   // (193,)
    float* out = (float*)d_out;               // (16, 193, 128, 128)

    const int blocks = NBATCH * (HW / 4 / TPB) * NCHUNK;  // 16*16*2 = 512
    neigh_conv_kernel<<<dim3(blocks), dim3(TPB), 0, stream>>>(x, W, b, out);
}